// Net_91164975824989
// MI455X (gfx1250) — compile-verified
//
#include <hip/hip_runtime.h>
#include <math.h>

// ---------------------------------------------------------------------------
// Strategy:
//  * Spikes are binary -> each 3-element "line" is one of 8 patterns.
//  * The whole MLP (3->30->30->3->2, ReLU, softmax) is evaluated ONCE on the
//    8 patterns using V_WMMA_F32_16X16X4_F32 (pure-fp32 CDNA5 WMMA), giving an
//    8-entry LUT of softmax pairs in LDS.
//  * Each thread then runs the 25-step LIF recurrence for one batch element
//    and accumulates 6 LUT lookups per step. Memory traffic = 1.4 MB total ->
//    this kernel is launch-latency bound on MI455X (23.3 TB/s).
// ---------------------------------------------------------------------------

typedef float v2f __attribute__((ext_vector_type(2)));
typedef float v8f __attribute__((ext_vector_type(8)));

#define BETA   0.95f
#define THRESH 1.0f
#define TSTEPS 25

__device__ inline v8f wmma4(v2f a, v2f b, v8f c) {
  // (neg_a, A, neg_b, B, c_mod, C, reuse_a, reuse_b)
  return __builtin_amdgcn_wmma_f32_16x16x4_f32(false, a, false, b, (short)0, c,
                                               false, false);
}

// B-tile (4x16, K x N) from row-major weight W[outDim][inDim]:
// Bmat[k][n] = W[n][k].  Lanes 0-15 hold column n = nbase+lane with K = 4kc,
// 4kc+1 in b.x/b.y; lanes 16-31 hold K = 4kc+2, 4kc+3. Zero-padded OOB.
__device__ inline v2f btile(const float* __restrict__ W, int outDim, int inDim,
                            int nbase, int kc, int lane) {
  int half = lane >> 4;
  int n    = nbase + (lane & 15);
  int k0   = 4 * kc + 2 * half;
  v2f b;
  b.x = (n < outDim && (k0 + 0) < inDim) ? W[n * inDim + k0 + 0] : 0.0f;
  b.y = (n < outDim && (k0 + 1) < inDim) ? W[n * inDim + k0 + 1] : 0.0f;
  return b;
}

// A-tile (16x4, M x K) from LDS activation stage S[8][32] (rows 0..7 valid,
// columns zero-padded). Rows 8..15 forced to zero.
__device__ inline v2f atile(const float (*S)[32], int kc, int lane) {
  int half = lane >> 4;
  int m    = lane & 15;
  int k0   = 4 * kc + 2 * half;
  v2f a;
  a.x = (m < 8) ? S[m][k0 + 0] : 0.0f;
  a.y = (m < 8) ? S[m][k0 + 1] : 0.0f;
  return a;
}

// C init = bias broadcast down each column (rows 8..15 are never read back).
__device__ inline v8f ctile(const float* __restrict__ bias, int outDim,
                            int nbase, int lane) {
  int   n  = nbase + (lane & 15);
  float bv = (n < outDim) ? bias[n] : 0.0f;
  v8f   c;
#pragma unroll
  for (int j = 0; j < 8; ++j) c[j] = bv;
  return c;
}

__global__ void __launch_bounds__(256)
snn_lut_kernel(const float* __restrict__ x,
               const float* __restrict__ W1, const float* __restrict__ b1,
               const float* __restrict__ W2, const float* __restrict__ b2,
               const float* __restrict__ W3, const float* __restrict__ b3,
               const float* __restrict__ W4, const float* __restrict__ b4,
               float* __restrict__ out, int Btotal) {
  __shared__ float  stage[2][8][32];  // activation staging (D-layout -> A-layout)
  __shared__ float  h4s[8][2];        // post-ReLU layer-4 logits
  __shared__ float2 lut[8];           // softmax LUT: pattern -> (p0, p1)

  const int lane = threadIdx.x & 31;

  // ---------------- LUT build: every wave computes it redundantly ----------
  // (WMMA needs EXEC all-1s; identical values make shared-mem races benign.)

  // Pattern matrix A0 (16x4): row m<8 = bits of pattern m, K=3 zero-padded.
  v2f ap;
  {
    int half = lane >> 4;
    int m    = lane & 15;
    int k0   = 2 * half;  // {0,1} or {2,3}
    ap.x = (m < 8 && (k0 + 0) < 3) ? (float)((m >> (k0 + 0)) & 1) : 0.0f;
    ap.y = (m < 8 && (k0 + 1) < 3) ? (float)((m >> (k0 + 1)) & 1) : 0.0f;
  }

  // Layer 1: [8x3] x [3x30] + b1, ReLU  (2 WMMAs)
#pragma unroll
  for (int t = 0; t < 2; ++t) {
    v8f c = ctile(b1, 30, 16 * t, lane);
    c     = wmma4(ap, btile(W1, 30, 3, 16 * t, 0, lane), c);
    if (lane < 16) {
#pragma unroll
      for (int j = 0; j < 8; ++j)
        stage[0][j][16 * t + lane] = fmaxf(c[j], 0.0f);
    }
  }
  __syncthreads();

  // Layer 2: [8x30] x [30x30] + b2, ReLU  (16 WMMAs)
#pragma unroll
  for (int t = 0; t < 2; ++t) {
    v8f c = ctile(b2, 30, 16 * t, lane);
#pragma unroll
    for (int kc = 0; kc < 8; ++kc)
      c = wmma4(atile(stage[0], kc, lane),
                btile(W2, 30, 30, 16 * t, kc, lane), c);
    if (lane < 16) {
#pragma unroll
      for (int j = 0; j < 8; ++j)
        stage[1][j][16 * t + lane] = fmaxf(c[j], 0.0f);
    }
  }
  __syncthreads();

  // Layer 3: [8x30] x [30x3] + b3, ReLU  (8 WMMAs, one N-tile)
  {
    v8f c = ctile(b3, 3, 0, lane);
#pragma unroll
    for (int kc = 0; kc < 8; ++kc)
      c = wmma4(atile(stage[1], kc, lane),
                btile(W3, 3, 30, 0, kc, lane), c);
    if (lane < 16) {
#pragma unroll
      for (int j = 0; j < 8; ++j) {
        stage[0][j][lane]      = fmaxf(c[j], 0.0f);
        stage[0][j][16 + lane] = 0.0f;  // zero the uncomputed tile
      }
    }
  }
  __syncthreads();

  // Layer 4: [8x3] x [3x2] + b4, ReLU  (1 WMMA)
  {
    v8f c = ctile(b4, 2, 0, lane);
    c     = wmma4(atile(stage[0], 0, lane), btile(W4, 2, 3, 0, 0, lane), c);
    // h4[p][o] lives in lane o (o<2), VGPR p (p<8)
    if (lane < 2) {
#pragma unroll
      for (int p = 0; p < 8; ++p) h4s[p][lane] = fmaxf(c[p], 0.0f);
    }
  }
  __syncthreads();

  // Softmax over the size-2 axis -> LUT (threads 0..7 of the block).
  if (threadIdx.x < 8) {
    float a0 = h4s[threadIdx.x][0];
    float a1 = h4s[threadIdx.x][1];
    float m  = fmaxf(a0, a1);
    float e0 = expf(a0 - m);
    float e1 = expf(a1 - m);
    float iv = 1.0f / (e0 + e1);
    lut[threadIdx.x] = make_float2(e0 * iv, e1 * iv);
  }
  __syncthreads();

  // ---------------- LIF scan + LUT accumulation ----------------------------
  int tid = blockIdx.x * blockDim.x + threadIdx.x;
  if (tid < Btotal) {
    const float* xp = x + (size_t)tid * 9;
    float xv[9], mem[9];
#pragma unroll
    for (int k = 0; k < 9; ++k) { xv[k] = xp[k]; mem[k] = 0.0f; }

    float acc0 = 0.0f, acc1 = 0.0f;
    for (int t = 0; t < TSTEPS; ++t) {
      unsigned sm = 0u;
#pragma unroll
      for (int k = 0; k < 9; ++k) {
        float r = (mem[k] > THRESH) ? THRESH : 0.0f;  // reset from prev mem
        mem[k]  = BETA * mem[k] + xv[k] - r;
        sm |= (mem[k] > THRESH) ? (1u << k) : 0u;
      }
      // 3 row lines: bits (3l, 3l+1, 3l+2)
      int i0 = sm & 7;
      int i1 = (sm >> 3) & 7;
      int i2 = (sm >> 6) & 7;
      // 3 column lines: bits (c, 3+c, 6+c)
      int c0 = ((sm >> 0) & 1) | (((sm >> 3) & 1) << 1) | (((sm >> 6) & 1) << 2);
      int c1 = ((sm >> 1) & 1) | (((sm >> 4) & 1) << 1) | (((sm >> 7) & 1) << 2);
      int c2 = ((sm >> 2) & 1) | (((sm >> 5) & 1) << 1) | (((sm >> 8) & 1) << 2);

      float2 p0 = lut[i0], p1 = lut[i1], p2 = lut[i2];
      float2 q0 = lut[c0], q1 = lut[c1], q2 = lut[c2];
      acc0 += p0.x + p1.x + p2.x + q0.x + q1.x + q2.x;
      acc1 += p0.y + p1.y + p2.y + q0.y + q1.y + q2.y;
    }
    out[(size_t)tid * 2 + 0] = acc0;
    out[(size_t)tid * 2 + 1] = acc1;
  }
}

extern "C" void kernel_launch(void* const* d_in, const int* in_sizes, int n_in,
                              void* d_out, int out_size, void* d_ws,
                              size_t ws_size, hipStream_t stream) {
  (void)n_in; (void)out_size; (void)d_ws; (void)ws_size;
  const float* x  = (const float*)d_in[0];
  const float* W1 = (const float*)d_in[1];
  const float* b1 = (const float*)d_in[2];
  const float* W2 = (const float*)d_in[3];
  const float* b2 = (const float*)d_in[4];
  const float* W3 = (const float*)d_in[5];
  const float* b3 = (const float*)d_in[6];
  const float* W4 = (const float*)d_in[7];
  const float* b4 = (const float*)d_in[8];
  float* out = (float*)d_out;

  int Btotal = in_sizes[0] / 9;  // x is [B, 3, 3]
  int block  = 256;              // 8 waves (wave32)
  int grid   = (Btotal + block - 1) / block;
  snn_lut_kernel<<<grid, block, 0, stream>>>(x, W1, b1, W2, b2, W3, b3, W4, b4,
                                             out, Btotal);
}